// NeighborhoodPool_1675037245698
// MI455X (gfx1250) — compile-verified
//
#include <hip/hip_runtime.h>

// ---------------------------------------------------------------------------
// NeighborhoodPool for MI455X (gfx1250, wave32).
//
// Pipeline (all on `stream`, graph-capture safe — no mallocs/syncs):
//   1) score_wmma      : sroot[n]=dot(feat,w_root), srel[n]=dot(feat,w_rel)
//                        via V_WMMA_F32_16X16X4_F32. Feature tiles staged
//                        global->LDS with GLOBAL_LOAD_ASYNC_TO_LDS_B32
//                        (ASYNCcnt); inner loop is branch-free:
//                        ds_load_b64(A) + ds_load_b64(B) + v_wmma per K-step.
//   2) k_score_init    : score = sroot + srel(self-loop) + b
//   3) k_score_edges   : score[dst] += srel[src]   (global_atomic_add_f32)
//   4) k_best_*        : per-node argmax-score over closed neighborhood
//                        (packed (score,id) u64 atomicMax), pointer jumping
//   5) blocked scan    : deterministic compaction of cluster centers (LDS)
//   6) pooling         : segment_max(x) via monotone-key u32 atomicMax,
//                        cross-cluster edge emission, center pos gather
// ---------------------------------------------------------------------------

typedef __attribute__((ext_vector_type(2))) float v2f;
typedef __attribute__((ext_vector_type(8))) float v8f;

#define F_DIM 125
#define P_DIM 3
#define IN_CH 128          // 125 + 3 : exactly 32 x (K=4) WMMA steps
#define TILE_NODES 64      // nodes per block (4 waves x 16-row WMMA tiles)
#define ROW_PITCH 132      // floats; 132*4 B is 8B-aligned, (4m+k0)%64 banks
                           // conflict-free, allows ds_load_b64 fragments

// ---- helpers --------------------------------------------------------------

// monotone float -> u32 key (order-preserving), and inverse
__device__ __forceinline__ unsigned int fkey(float f) {
  unsigned int u = __float_as_uint(f);
  return (u & 0x80000000u) ? ~u : (u | 0x80000000u);
}
__device__ __forceinline__ float fkey_dec(unsigned int k) {
  unsigned int u = (k & 0x80000000u) ? (k & 0x7FFFFFFFu) : ~k;
  return __uint_as_float(u);
}
__device__ __forceinline__ unsigned long long pack_sid(float s, int id) {
  return (((unsigned long long)fkey(s)) << 32) | (unsigned int)id;
}

// low 32 bits of a generic pointer to LDS == LDS byte address (aperture map)
__device__ __forceinline__ unsigned int lds_addr32(const void* p) {
  return (unsigned int)(unsigned long long)p;
}

// gfx1250 async copy: global -> LDS, one dword per lane, tracked by ASYNCcnt
__device__ __forceinline__ void async_g2l_b32(unsigned int lds_byte_addr,
                                              const void* gaddr) {
  asm volatile("global_load_async_to_lds_b32 %0, %1, off"
               :
               : "v"(lds_byte_addr), "v"((unsigned long long)gaddr)
               : "memory");
}
__device__ __forceinline__ void wait_async0() {
  asm volatile("s_wait_asynccnt 0x0" ::: "memory");
}

// ---- 1) WMMA scoring ------------------------------------------------------
// Block = 128 threads (4 waves) = 64 nodes. LDS: 64x128 feature tile plus a
// 16x128 combined weight tile (row 0 = w_root, row 1 = w_rel, rows 2..15 = 0)
// so the B fragment is a single unconditional ds_load_b64 per lane.
// A fragment (16x4 f32): lanes 0-15 rows M=0..15 at K={k0,k0+1},
// lanes 16-31 same rows at K={k0+2,k0+3}. D (16x16 f32): lane N==0 carries
// the w_root dots, lane N==1 the w_rel dots, rows M = v + 8*(lane>=16).
__global__ __launch_bounds__(128) void score_wmma(
    const float* __restrict__ x, const float* __restrict__ pos,
    const float* __restrict__ w_root, const float* __restrict__ w_rel,
    float* __restrict__ sroot, float* __restrict__ srel, int n) {
  __shared__ float sfeat[TILE_NODES * ROW_PITCH];   // 33.8 KB
  __shared__ float wcomb[16 * ROW_PITCH];           //  8.4 KB

  const int tid  = threadIdx.x;            // 0..127
  const int base = blockIdx.x * TILE_NODES;

  // --- stage combined weight tile (cold path, once per block) ---
  for (int idx = tid; idx < 16 * ROW_PITCH; idx += 128) {
    const int row = idx / ROW_PITCH, col = idx - row * ROW_PITCH;
    float v = 0.0f;
    if (row == 0 && col < IN_CH) v = w_root[col];
    if (row == 1 && col < IN_CH) v = w_rel[col];
    wcomb[idx] = v;
  }

  // --- stage x rows: async global->LDS, clamped node (no divergence) ---
  for (int idx = tid; idx < TILE_NODES * F_DIM; idx += 128) {
    const int ln = idx / F_DIM, k = idx - ln * F_DIM;
    const int node = min(base + ln, n - 1);
    async_g2l_b32(lds_addr32(&sfeat[ln * ROW_PITCH + k]),
                  &x[(long long)node * F_DIM + k]);
  }
  // --- stage pos into columns 125..127 ---
  for (int idx = tid; idx < TILE_NODES * P_DIM; idx += 128) {
    const int ln = idx / P_DIM, j = idx - ln * P_DIM;
    const int node = min(base + ln, n - 1);
    async_g2l_b32(lds_addr32(&sfeat[ln * ROW_PITCH + F_DIM + j]),
                  &pos[(long long)node * P_DIM + j]);
  }
  wait_async0();
  __syncthreads();

  const int lane = tid & 31;
  const int wv   = tid >> 5;               // wave id 0..3 -> 16-node tile
  const int m    = lane & 15;
  const bool hi  = lane >= 16;
  const int ksh  = hi ? 2 : 0;
  const float* __restrict__ arow = &sfeat[(wv * 16 + m) * ROW_PITCH];
  const float* __restrict__ wrow = &wcomb[m * ROW_PITCH];

  v8f c = {};
#pragma unroll
  for (int kc = 0; kc < 32; ++kc) {
    const int k0 = kc * 4 + ksh;
    const v2f a  = *(const v2f*)(arow + k0);   // ds_load_b64, conflict-free
    const v2f bm = *(const v2f*)(wrow + k0);   // ds_load_b64, branch-free B
    // 8 args: (neg_a, A, neg_b, B, c_mod, C, reuse_a, reuse_b)
    c = __builtin_amdgcn_wmma_f32_16x16x4_f32(
        false, a, false, bm, (short)0, c, false, false);
  }

  const int mrow = hi ? 8 : 0;
  if (m == 0) {
#pragma unroll
    for (int v = 0; v < 8; ++v) {
      const int nn = base + wv * 16 + mrow + v;
      if (nn < n) sroot[nn] = c[v];
    }
  } else if (m == 1) {
#pragma unroll
    for (int v = 0; v < 8; ++v) {
      const int nn = base + wv * 16 + mrow + v;
      if (nn < n) srel[nn] = c[v];
    }
  }
}

// ---- 2..3) score assembly -------------------------------------------------

__global__ void k_zero_counters(int* c) {
  if (threadIdx.x < 8) c[threadIdx.x] = 0;
}

__global__ void k_score_init(const float* __restrict__ sroot,
                             const float* __restrict__ srel,
                             const float* __restrict__ b,
                             float* __restrict__ score, int n) {
  int i = blockIdx.x * blockDim.x + threadIdx.x;
  if (i < n) score[i] = sroot[i] + srel[i] + b[0];  // self-loop included
}

__global__ void k_score_edges(const int* __restrict__ ei,
                              const float* __restrict__ srel,
                              float* __restrict__ score, int nE) {
  int e = blockIdx.x * blockDim.x + threadIdx.x;
  if (e >= nE) return;
  int s = ei[e], d = ei[nE + e];
  __builtin_prefetch(&srel[s], 0, 1);   // global_prefetch_b8
  if (s != d) atomicAdd(&score[d], srel[s]);
}

// ---- 4) greedy center selection (BFS approximation) -----------------------

__global__ void k_best_init(const float* __restrict__ score,
                            unsigned long long* __restrict__ best, int n) {
  int i = blockIdx.x * blockDim.x + threadIdx.x;
  if (i < n) best[i] = pack_sid(score[i], i);
}

__global__ void k_best_edges(const int* __restrict__ ei,
                             const float* __restrict__ score,
                             unsigned long long* __restrict__ best, int nE) {
  int e = blockIdx.x * blockDim.x + threadIdx.x;
  if (e >= nE) return;
  int s = ei[e], d = ei[nE + e];
  if (s == d) return;
  __builtin_prefetch(&score[s], 0, 1);
  atomicMax(&best[d], pack_sid(score[s], s));
  atomicMax(&best[s], pack_sid(score[d], d));   // undirected reachability
}

__global__ void k_hop(const unsigned long long* __restrict__ best,
                      int* __restrict__ label, int n) {
  int i = blockIdx.x * blockDim.x + threadIdx.x;
  if (i < n) label[i] = (int)(best[i] & 0xFFFFFFFFull);
}

__global__ void k_jump(const int* __restrict__ in, int* __restrict__ out, int n) {
  int i = blockIdx.x * blockDim.x + threadIdx.x;
  if (i < n) out[i] = in[in[i]];
}

// ---- 5) deterministic compaction (LDS reduce + scan) ----------------------

__global__ void k_mark(const int* __restrict__ label, int* __restrict__ isc, int n) {
  int i = blockIdx.x * blockDim.x + threadIdx.x;
  if (i < n) isc[i] = (label[i] == i) ? 1 : 0;
}

__global__ void k_block_count(const int* __restrict__ isc,
                              int* __restrict__ bsum, int n) {
  __shared__ int sm[256];
  int i = blockIdx.x * 256 + threadIdx.x;
  sm[threadIdx.x] = (i < n) ? isc[i] : 0;
  __syncthreads();
#pragma unroll
  for (int st = 128; st > 0; st >>= 1) {
    if (threadIdx.x < st) sm[threadIdx.x] += sm[threadIdx.x + st];
    __syncthreads();
  }
  if (threadIdx.x == 0) bsum[blockIdx.x] = sm[0];
}

__global__ void k_scan_serial(const int* __restrict__ bsum,
                              int* __restrict__ boff, int nb,
                              int* __restrict__ counters) {
  if (blockIdx.x == 0 && threadIdx.x == 0) {
    int acc = 0;
    for (int i = 0; i < nb; ++i) { boff[i] = acc; acc += bsum[i]; }
    counters[0] = acc;                        // c = number of clusters
  }
}

__global__ void k_assign_cid(const int* __restrict__ isc,
                             const int* __restrict__ boff,
                             int* __restrict__ cmap, int n) {
  __shared__ int sm[256];
  int i = blockIdx.x * 256 + threadIdx.x;
  int v = (i < n) ? isc[i] : 0;
  sm[threadIdx.x] = v;
  __syncthreads();
  for (int st = 1; st < 256; st <<= 1) {      // inclusive Hillis-Steele
    int t = (threadIdx.x >= st) ? sm[threadIdx.x - st] : 0;
    __syncthreads();
    sm[threadIdx.x] += t;
    __syncthreads();
  }
  if (i < n && v) cmap[i] = boff[blockIdx.x] + sm[threadIdx.x] - 1;
}

__global__ void k_cluster(const int* __restrict__ label,
                          const int* __restrict__ cmap,
                          int* __restrict__ cluster, int n) {
  int i = blockIdx.x * blockDim.x + threadIdx.x;
  if (i < n) cluster[i] = cmap[label[i]];
}

// ---- 6) pooling into d_out ------------------------------------------------
// x_p region [0, c*125) is built as monotone u32 keys, then decoded in place.

__global__ void k_init_xkeys(float* __restrict__ out, const int* __restrict__ counters,
                             int out_size) {
  long long idx = (long long)blockIdx.x * blockDim.x + threadIdx.x;
  long long lim = (long long)counters[0] * F_DIM;
  if (idx < lim && idx < out_size)
    ((unsigned int*)out)[idx] = 0x007FFFFFu;  // key(-inf)
}

__global__ void k_pool_x(const float* __restrict__ x,
                         const int* __restrict__ cluster,
                         float* __restrict__ out, int out_size, int n) {
  long long idx = (long long)blockIdx.x * blockDim.x + threadIdx.x;
  if (idx >= (long long)n * F_DIM) return;
  int node = (int)(idx / F_DIM), f = (int)(idx % F_DIM);
  long long dst = (long long)cluster[node] * F_DIM + f;
  if (dst < out_size)
    atomicMax((unsigned int*)out + dst, fkey(x[idx]));   // segment_max
}

__global__ void k_decode_x(float* __restrict__ out, const int* __restrict__ counters,
                           int out_size) {
  long long idx = (long long)blockIdx.x * blockDim.x + threadIdx.x;
  long long lim = (long long)counters[0] * F_DIM;
  if (idx < lim && idx < out_size)
    out[idx] = fkey_dec(((unsigned int*)out)[idx]);
}

__global__ void k_count_edges(const int* __restrict__ ei,
                              const int* __restrict__ cluster,
                              int* __restrict__ counters, int nE) {
  int e = blockIdx.x * blockDim.x + threadIdx.x;
  if (e >= nE) return;
  int cs = cluster[ei[e]], cd = cluster[ei[nE + e]];
  if (cs != cd) atomicAdd(&counters[1], 1);
}

__global__ void k_emit_edges(const int* __restrict__ ei,
                             const int* __restrict__ cluster,
                             float* __restrict__ out,
                             int* __restrict__ counters, int out_size, int nE) {
  int e = blockIdx.x * blockDim.x + threadIdx.x;
  if (e >= nE) return;
  int cs = cluster[ei[e]], cd = cluster[ei[nE + e]];
  if (cs == cd) return;
  int idx = atomicAdd(&counters[2], 1);
  long long base = (long long)counters[0] * F_DIM;
  long long Ep   = counters[1];
  if (base + idx < out_size)      out[base + idx]      = (float)cs;
  if (base + Ep + idx < out_size) out[base + Ep + idx] = (float)cd;
}

__global__ void k_pos(const float* __restrict__ pos,
                      const int* __restrict__ isc,
                      const int* __restrict__ cmap,
                      float* __restrict__ out,
                      const int* __restrict__ counters, int out_size, int n) {
  int i = blockIdx.x * blockDim.x + threadIdx.x;
  if (i >= n || !isc[i]) return;
  long long base = (long long)counters[0] * F_DIM + 2LL * counters[1];
  long long o = base + (long long)cmap[i] * P_DIM;
#pragma unroll
  for (int j = 0; j < P_DIM; ++j)
    if (o + j < out_size) out[o + j] = pos[i * P_DIM + j];
}

// ---------------------------------------------------------------------------

extern "C" void kernel_launch(void* const* d_in, const int* in_sizes, int n_in,
                              void* d_out, int out_size, void* d_ws, size_t ws_size,
                              hipStream_t stream) {
  const float* x      = (const float*)d_in[0];
  const int*   ei     = (const int*)  d_in[1];
  const float* pos    = (const float*)d_in[2];
  const float* w_root = (const float*)d_in[3];
  const float* w_rel  = (const float*)d_in[4];
  const float* b      = (const float*)d_in[5];

  const int N  = in_sizes[0] / F_DIM;
  const int E  = in_sizes[1] / 2;
  const int nb = (N + 255) / 256;

  // workspace carve-out (8-byte aligned slices)
  char*  ws  = (char*)d_ws;
  size_t off = 0;
  auto carve = [&](size_t bytes) {
    void* p = ws + off;
    off = (off + bytes + 7) & ~(size_t)7;
    return p;
  };
  unsigned long long* best = (unsigned long long*)carve((size_t)N * 8);
  float* sroot   = (float*)carve((size_t)N * 4);
  float* srel    = (float*)carve((size_t)N * 4);
  float* score   = (float*)carve((size_t)N * 4);
  int*   label   = (int*)  carve((size_t)N * 4);
  int*   label2  = (int*)  carve((size_t)N * 4);
  int*   isc     = (int*)  carve((size_t)N * 4);
  int*   cmap    = (int*)  carve((size_t)N * 4);
  int*   cluster = (int*)  carve((size_t)N * 4);
  int*   bsum    = (int*)  carve((size_t)nb * 4);
  int*   boff    = (int*)  carve((size_t)nb * 4);
  int*   cnt     = (int*)  carve(8 * 4);
  (void)ws_size; (void)n_in;

  float* out = (float*)d_out;
  const int nodeGrid = nb;
  const int edgeGrid = (E + 255) / 256;
  const long long nxf = (long long)N * F_DIM;
  const int xGrid = (int)((nxf + 255) / 256);

  k_zero_counters<<<1, 32, 0, stream>>>(cnt);

  // (1) WMMA scoring: 4 waves / 64 nodes per block, async LDS staging
  score_wmma<<<(N + TILE_NODES - 1) / TILE_NODES, 128, 0, stream>>>(
      x, pos, w_root, w_rel, sroot, srel, N);

  // (2,3) score = self terms + edge scatter
  k_score_init <<<nodeGrid, 256, 0, stream>>>(sroot, srel, b, score, N);
  k_score_edges<<<edgeGrid, 256, 0, stream>>>(ei, srel, score, E);

  // (4) center selection + pointer jumping
  k_best_init <<<nodeGrid, 256, 0, stream>>>(score, best, N);
  k_best_edges<<<edgeGrid, 256, 0, stream>>>(ei, score, best, E);
  k_hop       <<<nodeGrid, 256, 0, stream>>>(best, label, N);
  k_jump      <<<nodeGrid, 256, 0, stream>>>(label, label2, N);
  k_jump      <<<nodeGrid, 256, 0, stream>>>(label2, label, N);

  // (5) deterministic compaction
  k_mark       <<<nodeGrid, 256, 0, stream>>>(label, isc, N);
  k_block_count<<<nb, 256, 0, stream>>>(isc, bsum, N);
  k_scan_serial<<<1, 1, 0, stream>>>(bsum, boff, nb, cnt);
  k_assign_cid <<<nb, 256, 0, stream>>>(isc, boff, cmap, N);
  k_cluster    <<<nodeGrid, 256, 0, stream>>>(label, cmap, cluster, N);

  // (6) pooled outputs
  k_init_xkeys <<<xGrid, 256, 0, stream>>>(out, cnt, out_size);
  k_pool_x     <<<xGrid, 256, 0, stream>>>(x, cluster, out, out_size, N);
  k_decode_x   <<<xGrid, 256, 0, stream>>>(out, cnt, out_size);
  k_count_edges<<<edgeGrid, 256, 0, stream>>>(ei, cluster, cnt, E);
  k_emit_edges <<<edgeGrid, 256, 0, stream>>>(ei, cluster, out, cnt, out_size, E);
  k_pos        <<<nodeGrid, 256, 0, stream>>>(pos, isc, cmap, out, cnt, out_size, N);
}